// Ranker_49031346651809
// MI455X (gfx1250) — compile-verified
//
#include <hip/hip_runtime.h>
#include <hip/hip_bf16.h>
#include <math.h>
#include <cstdint>

// Problem constants (match reference)
namespace {
constexpr int B  = 2048;     // rows
constexpr int C  = 100000;   // classes
constexpr int L  = 50;       // labels per row (true item = last)
constexpr int K  = 29;       // negatives per row
constexpr int NC = 1 + K;    // 30 candidates
constexpr int WAVE = 32;
constexpr int WAVES_PER_BLOCK = 8;
constexpr int BLOCK = WAVE * WAVES_PER_BLOCK;   // 256 threads
constexpr int GRID  = B / WAVES_PER_BLOCK;      // 256 blocks
constexpr int NOUT  = 8;     // recall@1,ndcg@1,recall@5,ndcg@5,recall@10,ndcg@10,mrr,loss
}

// One wave32 per row: lanes 0..29 hold the 30 candidate scores.
// Scores are gathered via the CDNA5 async global->LDS path (ASYNCcnt),
// then reduced with wave32 shfl/ballot. Deterministic: fixed-order
// block reduction, no float atomics.
__global__ __launch_bounds__(BLOCK)
void ranker_rows_kernel(const float* __restrict__ scores,
                        const int*   __restrict__ labels,
                        const int*   __restrict__ neg,
                        float*       __restrict__ partials) {
    __shared__ float stage[WAVES_PER_BLOCK][WAVE];
    __shared__ float acc[WAVES_PER_BLOCK][NOUT];

    const int lane = threadIdx.x & (WAVE - 1);
    const int wv   = threadIdx.x >> 5;
    const int row  = blockIdx.x * WAVES_PER_BLOCK + wv;

    // Candidate class index for this lane.
    int cand;
    if (lane == 0) {
        cand = labels[row * L + (L - 1)];          // true item
    } else {
        int j = lane - 1;
        if (j >= K) j = 0;                         // lanes 30/31: valid dummy
        cand = neg[row * K + j];
    }

    const float* src = scores + (size_t)row * (size_t)C + (size_t)cand;

    // Per-lane LDS byte address == low 32 bits of the generic LDS pointer
    // (flat aperture: LDS offset lives in addr[31:0]).
    unsigned lds_addr = (unsigned)(size_t)&stage[wv][lane];

    // CDNA5 async gather: memory -> LDS, tracked by ASYNCcnt.
    asm volatile("global_load_async_to_lds_b32 %0, %1, off"
                 :: "v"(lds_addr), "v"(src)
                 : "memory");
    asm volatile("s_wait_asynccnt 0" ::: "memory");

    float s = stage[wv][lane];
    if (lane >= NC) s = -INFINITY;                 // mask dummy lanes

    // Wave max (log-softmax stabilization).
    float m = s;
    #pragma unroll
    for (int off = 16; off > 0; off >>= 1)
        m = fmaxf(m, __shfl_xor(m, off, WAVE));

    // Wave sum of exp(s - m).
    float e = (lane < NC) ? expf(s - m) : 0.0f;
    float sum = e;
    #pragma unroll
    for (int off = 16; off > 0; off >>= 1)
        sum += __shfl_xor(sum, off, WAVE);

    // True-item score broadcast + rank (stable-argsort semantics:
    // rank = #candidates strictly greater than the true score).
    float s0 = __shfl(s, 0, WAVE);
    unsigned long long gt = __ballot(s > s0);      // lanes >= NC are -inf -> false
    float r = (float)__popcll(gt);

    if (lane == 0) {
        float invl = 1.0f / log2f(r + 2.0f);
        float h1  = (r < 1.0f)  ? 1.0f : 0.0f;
        float h5  = (r < 5.0f)  ? 1.0f : 0.0f;
        float h10 = (r < 10.0f) ? 1.0f : 0.0f;
        acc[wv][0] = h1;            // recall@1
        acc[wv][1] = h1  * invl;    // ndcg@1
        acc[wv][2] = h5;            // recall@5
        acc[wv][3] = h5  * invl;    // ndcg@5
        acc[wv][4] = h10;           // recall@10
        acc[wv][5] = h10 * invl;    // ndcg@10
        acc[wv][6] = 1.0f / (r + 1.0f);            // mrr
        acc[wv][7] = -(s0 - m - logf(sum));        // CE loss contribution
    }
    __syncthreads();

    // Fixed-order per-block reduction (deterministic).
    if (threadIdx.x < NOUT) {
        float t = 0.0f;
        #pragma unroll
        for (int w = 0; w < WAVES_PER_BLOCK; ++w)
            t += acc[w][threadIdx.x];
        partials[blockIdx.x * NOUT + threadIdx.x] = t;
    }
}

// Single-block fixed-order reduction of the per-block partials -> means.
__global__ __launch_bounds__(WAVE)
void ranker_finalize_kernel(const float* __restrict__ partials,
                            float*       __restrict__ out) {
    int c = threadIdx.x;
    if (c < NOUT) {
        float t = 0.0f;
        for (int b = 0; b < GRID; ++b)
            t += partials[b * NOUT + c];
        out[c] = t * (1.0f / (float)B);
    }
}

extern "C" void kernel_launch(void* const* d_in, const int* in_sizes, int n_in,
                              void* d_out, int out_size, void* d_ws, size_t ws_size,
                              hipStream_t stream) {
    const float* scores = (const float*)d_in[0];   // [B, C] f32
    const int*   labels = (const int*)  d_in[1];   // [B, L] i32
    const int*   neg    = (const int*)  d_in[2];   // [B, K] i32
    float* out      = (float*)d_out;               // [8] f32
    float* partials = (float*)d_ws;                // GRID*NOUT floats (8 KB)

    ranker_rows_kernel<<<GRID, BLOCK, 0, stream>>>(scores, labels, neg, partials);
    ranker_finalize_kernel<<<1, WAVE, 0, stream>>>(partials, out);
}